// Generator_90950227460153
// MI455X (gfx1250) — compile-verified
//
#include <hip/hip_runtime.h>
#include <math.h>

// ---------------------------------------------------------------------------
// CDNA5 (gfx1250) WMMA: V_WMMA_F32_16X16X4_F32, full f32 precision.
// A: 16x4 f32 (2 VGPRs), B: 4x16 f32 (2 VGPRs), C/D: 16x16 f32 (8 VGPRs).
// Layout (ISA 7.12.2): A lane=M, VGPR+lanehalf=K; B lane=N, VGPR+lanehalf=K;
// C/D VGPR r: M = r + 8*(lane/16), N = lane&15.
// ---------------------------------------------------------------------------
typedef float v2f __attribute__((ext_vector_type(2)));
typedef float v8f __attribute__((ext_vector_type(8)));

__device__ __forceinline__ v8f wmma_f32_k4(v2f a, v2f b, v8f c) {
  // 8 args: (neg_a, A, neg_b, B, c_mod, C, reuse_a, reuse_b)
  return __builtin_amdgcn_wmma_f32_16x16x4_f32(
      false, a, false, b, (short)0, c, false, false);
}

__device__ __forceinline__ float sigmoidf_(float x) {
  return 1.0f / (1.0f + expf(-x));
}

// ---------------------------------------------------------------------------
// Zero-fill (accumulators + d_out must be cleared every call)
// ---------------------------------------------------------------------------
__global__ void zero_kernel(float* __restrict__ p, int n) {
  int i = blockIdx.x * blockDim.x + threadIdx.x;
  if (i < n) p[i] = 0.0f;
}

// ---------------------------------------------------------------------------
// OUT[n,o] = sigmoid( sum_k X[n,k] * W[o,k] + bias[o] ),  W row-major [OUTC,K].
// One wave computes a 16 x (16*CT) strip: CT independent accumulators share
// the A operand (pipelines WMMAs, 4x less A traffic).
// grid = (N/16, OUTC/(16*CT)), block = 32.
// ---------------------------------------------------------------------------
template <int K, int OUTC, int CT>
__global__ void gemm_xwT_sigmoid(const float* __restrict__ X,
                                 const float* __restrict__ W,
                                 const float* __restrict__ bias,
                                 float* __restrict__ OUT) {
  const int lane = threadIdx.x;
  const int half = lane >> 4;
  const int l16  = lane & 15;
  const int row0 = blockIdx.x * 16;
  const int col0 = blockIdx.y * (16 * CT);
  const float* xr = X + (size_t)(row0 + l16) * K + 2 * half;
  const float* wr[CT];
#pragma unroll
  for (int t = 0; t < CT; ++t)
    wr[t] = W + (size_t)(col0 + t * 16 + l16) * K + 2 * half;
  v8f acc[CT];
#pragma unroll
  for (int t = 0; t < CT; ++t) acc[t] = (v8f){};
#pragma unroll 8
  for (int k0 = 0; k0 < K; k0 += 4) {
    v2f a = *(const v2f*)(xr + k0);
#pragma unroll
    for (int t = 0; t < CT; ++t) {
      v2f b = *(const v2f*)(wr[t] + k0);
      acc[t] = wmma_f32_k4(a, b, acc[t]);
    }
  }
#pragma unroll
  for (int t = 0; t < CT; ++t) {
    const int col = col0 + t * 16 + l16;
    const float bcol = bias[col];
#pragma unroll
    for (int r = 0; r < 8; ++r) {
      const int m = r + 8 * half;
      OUT[(size_t)(row0 + m) * OUTC + col] = sigmoidf_(acc[t][r] + bcol);
    }
  }
}

// ---------------------------------------------------------------------------
// NNConv edge kernel: per-edge weight matrix w = relu(ea*fcw + fcb) [InC,OutC],
// msg = h[src] @ w, scatter-add into S[dst], count edges per dst node.
// One block (OutC threads) per edge.  (Batched GEMV with per-edge B: VALU.)
// ---------------------------------------------------------------------------
template <int INC>
__global__ void nnconv_edge(const float* __restrict__ H,
                            const int* __restrict__ src,
                            const int* __restrict__ dst,
                            const float* __restrict__ ea,
                            const float* __restrict__ fcw,
                            const float* __restrict__ fcb,
                            float* __restrict__ S,
                            float* __restrict__ cnt,
                            int OutC, int do_cnt) {
  __shared__ float hs[INC];
  const int e = blockIdx.x;
  const int t = threadIdx.x;
  const int se = src[e];
  const int de = dst[e];
  if (t < INC) hs[t] = H[(size_t)se * INC + t];
  __syncthreads();
  const float a = ea[e];
  float acc = 0.0f;
#pragma unroll 4
  for (int i = 0; i < INC; ++i) {
    const int j = i * OutC + t;              // reshape(-1, in, out) row-major
    float w = fmaf(a, fcw[j], fcb[j]);
    w = fmaxf(w, 0.0f);                       // ReLU
    acc = fmaf(hs[i], w, acc);
  }
  atomicAdd(&S[(size_t)de * OutC + t], acc);
  if (do_cnt && t == 0) atomicAdd(&cnt[de], 1.0f);
}

// ---------------------------------------------------------------------------
// Y[n,o] = S[n,o]/max(cnt[n],1) + (H @ root)[n,o] + bias[o]
// (root row-major [K,OUTC]).  Accumulates per-column sum/sumsq for BN.
// One wave per 16 x (16*CT) strip.  grid = (N/16, OUTC/(16*CT)), block = 32.
// ---------------------------------------------------------------------------
template <int K, int OUTC, int CT>
__global__ void root_agg_bnpartial(const float* __restrict__ H,
                                   const float* __restrict__ root,
                                   const float* __restrict__ bias,
                                   const float* __restrict__ S,
                                   const float* __restrict__ cnt,
                                   float* __restrict__ Y,
                                   float* __restrict__ colsum,
                                   float* __restrict__ colsq) {
  const int lane = threadIdx.x;
  const int half = lane >> 4;
  const int l16  = lane & 15;
  const int row0 = blockIdx.x * 16;
  const int col0 = blockIdx.y * (16 * CT);
  const float* xr = H + (size_t)(row0 + l16) * K + 2 * half;
  v8f acc[CT];
#pragma unroll
  for (int t = 0; t < CT; ++t) acc[t] = (v8f){};
#pragma unroll 4
  for (int k0 = 0; k0 < K; k0 += 4) {
    v2f a = *(const v2f*)(xr + k0);
    const float* r0 = root + (size_t)(k0 + 2 * half) * OUTC + col0 + l16;
#pragma unroll
    for (int t = 0; t < CT; ++t) {
      v2f b;
      b.x = r0[t * 16];
      b.y = r0[t * 16 + OUTC];
      acc[t] = wmma_f32_k4(a, b, acc[t]);
    }
  }
#pragma unroll
  for (int t = 0; t < CT; ++t) {
    const int col = col0 + t * 16 + l16;
    const float bcol = bias[col];
    float lsum = 0.0f, lsq = 0.0f;
#pragma unroll
    for (int r = 0; r < 8; ++r) {
      const int row = row0 + r + 8 * half;
      const float c = fmaxf(cnt[row], 1.0f);
      const float y = acc[t][r] + S[(size_t)row * OUTC + col] / c + bcol;
      Y[(size_t)row * OUTC + col] = y;
      lsum += y;
      lsq  = fmaf(y, y, lsq);
    }
    atomicAdd(&colsum[col], lsum);
    atomicAdd(&colsq[col], lsq);
  }
}

// ---------------------------------------------------------------------------
// BN finalize: scale/shift per column (biased variance, eps=1e-3)
// ---------------------------------------------------------------------------
__global__ void bn_finalize(const float* __restrict__ colsum,
                            const float* __restrict__ colsq,
                            const float* __restrict__ g,
                            const float* __restrict__ b,
                            float* __restrict__ scale,
                            float* __restrict__ shift,
                            int OutC, float invN) {
  int c = blockIdx.x * blockDim.x + threadIdx.x;
  if (c < OutC) {
    float m = colsum[c] * invN;
    float v = colsq[c] * invN - m * m;
    float sc = g[c] * rsqrtf(v + 1e-3f);
    scale[c] = sc;
    shift[c] = b[c] - m * sc;
  }
}

// ---------------------------------------------------------------------------
// In-place: Y = sigmoid(Y*scale[col] + shift[col])  (OutC power of two)
// ---------------------------------------------------------------------------
__global__ void bn_apply_sigmoid(float* __restrict__ Y,
                                 const float* __restrict__ scale,
                                 const float* __restrict__ shift,
                                 int total, int cmask) {
  int i = blockIdx.x * blockDim.x + threadIdx.x;
  if (i < total) {
    int c = i & cmask;
    Y[i] = sigmoidf_(fmaf(Y[i], scale[c], shift[c]));
  }
}

// ---------------------------------------------------------------------------
// OUT[C,C] += Xo^T @ Xo over a K-chunk.  One wave: 16 x (16*CT) strip.
// grid = (C/16, C/(16*CT), nChunks), block = 32.  A shared across CT tiles.
// ---------------------------------------------------------------------------
template <int C, int CT>
__global__ void ata_kernel(const float* __restrict__ Xo,
                           float* __restrict__ OUT,
                           int N, int kchunk) {
  const int lane = threadIdx.x;
  const int half = lane >> 4;
  const int l16  = lane & 15;
  const int row0 = blockIdx.x * 16;           // output row = column of Xo (A)
  const int col0 = blockIdx.y * (16 * CT);
  const int kbeg = blockIdx.z * kchunk;
  int kend = kbeg + kchunk;
  if (kend > N) kend = N;
  const int ca = row0 + l16;
  v8f acc[CT];
#pragma unroll
  for (int t = 0; t < CT; ++t) acc[t] = (v8f){};
#pragma unroll 4
  for (int k0 = kbeg; k0 + 3 < kend; k0 += 4) {
    const size_t kb = (size_t)(k0 + 2 * half) * C;
    v2f a;
    a.x = Xo[kb + ca];
    a.y = Xo[kb + C + ca];
#pragma unroll
    for (int t = 0; t < CT; ++t) {
      const int cb = col0 + t * 16 + l16;
      v2f b;
      b.x = Xo[kb + cb];
      b.y = Xo[kb + C + cb];
      acc[t] = wmma_f32_k4(a, b, acc[t]);
    }
  }
#pragma unroll
  for (int t = 0; t < CT; ++t) {
#pragma unroll
    for (int r = 0; r < 8; ++r) {
      const int m = r + 8 * half;
      atomicAdd(&OUT[(size_t)(row0 + m) * C + col0 + t * 16 + l16], acc[t][r]);
    }
  }
}

// ---------------------------------------------------------------------------
extern "C" void kernel_launch(void* const* d_in, const int* in_sizes, int n_in,
                              void* d_out, int out_size, void* d_ws, size_t ws_size,
                              hipStream_t stream) {
  const float* x     = (const float*)d_in[0];
  const int*   ei    = (const int*)d_in[1];
  const float* ea    = (const float*)d_in[2];
  const float* piw   = (const float*)d_in[3];
  const float* pib   = (const float*)d_in[4];
  const float* fc1w  = (const float*)d_in[5];
  const float* fc1b  = (const float*)d_in[6];
  const float* root1 = (const float*)d_in[7];
  const float* bias1 = (const float*)d_in[8];
  const float* bn1g  = (const float*)d_in[9];
  const float* bn1b  = (const float*)d_in[10];
  const float* fc3w  = (const float*)d_in[11];
  const float* fc3b  = (const float*)d_in[12];
  const float* root3 = (const float*)d_in[13];
  const float* bias3 = (const float*)d_in[14];
  const float* bn3g  = (const float*)d_in[15];
  const float* bn3b  = (const float*)d_in[16];
  const float* pow_  = (const float*)d_in[17];
  const float* pob   = (const float*)d_in[18];

  const int NSRC = 256, H1 = 32, H2 = 64, H3 = 128, NTGT = 256;
  const int N = in_sizes[0] / NSRC;     // 10000
  const int E = in_sizes[1] / 2;        // 20000
  const int* src = ei;
  const int* dst = ei + E;

  // ---- workspace layout (floats) ----
  float* W = (float*)d_ws;
  size_t off = 0;
  float* h0 = W + off; off += (size_t)N * H1;      // sigmoid proj_in
  float* x1 = W + off; off += (size_t)N * H2;      // y1 then x1 (in place)
  float* x3 = W + off; off += (size_t)N * H3;      // y3 then x3 (in place)
  float* xo = W + off; off += (size_t)N * NTGT;    // sigmoid proj_out
  float* accRegion = W + off;                      // ---- zeroed each call ----
  float* s1  = W + off; off += (size_t)N * H2;
  float* s3  = W + off; off += (size_t)N * H3;
  float* cnt = W + off; off += (size_t)N;
  float* cs1 = W + off; off += H2;
  float* cq1 = W + off; off += H2;
  float* cs3 = W + off; off += H3;
  float* cq3 = W + off; off += H3;
  const int accN = (int)((W + off) - accRegion);   // ---- end zero region ----
  float* scale1 = W + off; off += H2;
  float* shift1 = W + off; off += H2;
  float* scale3 = W + off; off += H3;
  float* shift3 = W + off; off += H3;
  (void)ws_size; (void)n_in;

  const float invN = 1.0f / (float)N;
  float* out = (float*)d_out;

  // 0) clear accumulators + output
  zero_kernel<<<(accN + 255) / 256, 256, 0, stream>>>(accRegion, accN);
  zero_kernel<<<(out_size + 255) / 256, 256, 0, stream>>>(out, out_size);

  // 1) h0 = sigmoid(x @ proj_in_w^T + b)          [N,256] -> [N,32]
  gemm_xwT_sigmoid<256, 32, 2><<<dim3(N / 16, 1), 32, 0, stream>>>(
      x, piw, pib, h0);

  // 2) conv1 edge messages -> s1, counts -> cnt
  nnconv_edge<32><<<E, H2, 0, stream>>>(h0, src, dst, ea, fc1w, fc1b,
                                        s1, cnt, H2, 1);

  // 3) y1 = mean-agg + h0@root1 + bias1; BN partial sums   (K=32, OutC=64)
  root_agg_bnpartial<32, 64, 4><<<dim3(N / 16, 1), 32, 0, stream>>>(
      h0, root1, bias1, s1, cnt, x1, cs1, cq1);
  bn_finalize<<<1, H2, 0, stream>>>(cs1, cq1, bn1g, bn1b, scale1, shift1, H2, invN);
  bn_apply_sigmoid<<<((N * H2) + 255) / 256, 256, 0, stream>>>(
      x1, scale1, shift1, N * H2, H2 - 1);

  // 4) conv2 edge messages -> s3 (reuse cnt: same dst distribution)
  nnconv_edge<64><<<E, H3, 0, stream>>>(x1, src, dst, ea, fc3w, fc3b,
                                        s3, cnt, H3, 0);

  // 5) y3 = mean-agg + x1@root3 + bias3; BN partial sums   (K=64, OutC=128)
  root_agg_bnpartial<64, 128, 4><<<dim3(N / 16, 2), 32, 0, stream>>>(
      x1, root3, bias3, s3, cnt, x3, cs3, cq3);
  bn_finalize<<<1, H3, 0, stream>>>(cs3, cq3, bn3g, bn3b, scale3, shift3, H3, invN);
  bn_apply_sigmoid<<<((N * H3) + 255) / 256, 256, 0, stream>>>(
      x3, scale3, shift3, N * H3, H3 - 1);

  // 6) xo = sigmoid(x3 @ proj_out_w^T + b)        [N,128] -> [N,256]
  gemm_xwT_sigmoid<128, 256, 4><<<dim3(N / 16, 4), 32, 0, stream>>>(
      x3, pow_, pob, xo);

  // 7) out = xo^T @ xo                            [256,256]
  const int nchunk = 20;
  const int kchunk = (N + nchunk - 1) / nchunk;    // 500, multiple of 4
  ata_kernel<256, 4><<<dim3(16, 4, nchunk), 32, 0, stream>>>(
      xo, out, N, kchunk);
}